// FDS_35983236006151
// MI455X (gfx1250) — compile-verified
//
#include <hip/hip_runtime.h>

#define DIMS   512
#define NBINS  50
#define MPAD   64
#define NTOT   262144
#define KCHUNK 4096
#define TROWS  128                  // rows per TDM tile (KCHUNK % TROWS == 0)
#define OUT_NM 0
#define OUT_NV (NBINS*DIMS)
#define OUT_NN (2*NBINS*DIMS)
#define OUT_SM (2*NBINS*DIMS + NBINS)
#define OUT_SV (2*NBINS*DIMS + NBINS + NBINS*DIMS)

typedef __attribute__((ext_vector_type(2))) float        v2f;
typedef __attribute__((ext_vector_type(8))) float        v8f;
typedef __attribute__((ext_vector_type(4))) unsigned int v4u;
typedef __attribute__((ext_vector_type(4))) int          v4i;
typedef __attribute__((ext_vector_type(8))) int          v8i;

#if defined(__has_builtin)
#if __has_builtin(__builtin_amdgcn_tensor_load_to_lds)
#define HAVE_TDM 1
#endif
#endif
#ifndef HAVE_TDM
#define HAVE_TDM 0
#endif

#if HAVE_TDM
__device__ __forceinline__ void tdm_load(v4u g0, v8i g1) {
    v4i gz = {0, 0, 0, 0};
#if __clang_major__ >= 23
    v8i gz8 = {0, 0, 0, 0, 0, 0, 0, 0};
    __builtin_amdgcn_tensor_load_to_lds(g0, g1, gz, gz, gz8, 0);
#else
    __builtin_amdgcn_tensor_load_to_lds(g0, g1, gz, gz, 0);
#endif
}
#endif

// ---------------- workspace zero ----------------
__global__ void fds_zero(float* ws, int n) {
    int i = blockIdx.x * blockDim.x + threadIdx.x;
    if (i < n) ws[i] = 0.0f;
}

// ---------------- counts histogram ----------------
__global__ void fds_counts(const int* __restrict__ labels, float* __restrict__ cnt) {
    __shared__ unsigned int h[MPAD];
    int t = threadIdx.x;
    if (t < MPAD) h[t] = 0u;
    __syncthreads();
    int stride = gridDim.x * blockDim.x;
    for (int i = blockIdx.x * blockDim.x + t; i < NTOT; i += stride) {
        int b = labels[i] & (MPAD - 1);
        atomicAdd(&h[b], 1u);
    }
    __syncthreads();
    if (t < MPAD && h[t] != 0u) atomicAdd(&cnt[t], (float)h[t]);
}

// ---------------- WMMA one-hot segmented GEMM ----------------
// C_sum[64,512] += OneHot(labels)^T x features ; C_sq += OneHot^T x features^2
// TDM double-buffers 16-col x 128-row f32 tiles into LDS while the wave runs
// exact-f32 v_wmma_f32_16x16x4_f32 out of LDS. Features read from HBM once.
__global__ __launch_bounds__(32) void fds_wmma_reduce(
        const float* __restrict__ feat,
        const int*   __restrict__ labels,
        float* __restrict__ sums,
        float* __restrict__ sumsq) {
    __shared__ float smem[2 * TROWS * 16];      // 16 KB double buffer

    const int lane  = threadIdx.x;              // wave32
    const int ntile = blockIdx.x;               // 0..31  (512/16)
    const int kc    = blockIdx.y;               // 0..63  (262144/4096)
    const int l15   = lane & 15;
    const int khalf = (lane >> 4) << 1;         // K slot per lane-half: 0 or 2
    const int ncol  = ntile * 16 + l15;

    const v8f vzero = {0.f, 0.f, 0.f, 0.f, 0.f, 0.f, 0.f, 0.f};
    v8f accS[4] = {vzero, vzero, vzero, vzero};
    v8f accQ[4] = {vzero, vzero, vzero, vzero};

#if HAVE_TDM
    // D# group 1 (constant): data_size=4B, tensor 16 x TROWS, tile 16 x TROWS,
    // row stride = DIMS elements. 2-D tile => groups 2/3 zero, tile_dim2=0.
    v8i g1;
    g1[0] = 0x20000;          // data_size = 2 (4 bytes)
    g1[1] = 16 << 16;         // tensor_dim0 = 16
    g1[2] = TROWS << 16;      // tensor_dim1 = TROWS
    g1[3] = 16 << 16;         // tile_dim0 = 16
    g1[4] = TROWS;            // tile_dim1 = TROWS, tile_dim2 = 0
    g1[5] = DIMS;             // tensor_dim0_stride = 512
    g1[6] = 0;
    g1[7] = 0;

    const unsigned lds_base = (unsigned)(size_t)(void*)smem;   // LDS offset
    const unsigned long long gbase =
        (unsigned long long)(size_t)feat +
        ((unsigned long long)kc * KCHUNK * DIMS + (unsigned)(ntile * 16)) * 4ull;

    #define TDM_ISSUE(it, buf)                                                  \
        do {                                                                    \
            unsigned long long ga =                                             \
                gbase + (unsigned long long)(it) * (TROWS * DIMS * 4ull);       \
            v4u g0;                                                             \
            g0[0] = 1u;                               /* count = 1 */           \
            g0[1] = lds_base + (unsigned)(buf) * (TROWS * 16 * 4);              \
            g0[2] = (unsigned)(ga & 0xFFFFFFFFull);                             \
            g0[3] = (unsigned)((ga >> 32) & 0x01FFFFFFull) | (2u << 30);        \
            tdm_load(g0, g1);                                                   \
        } while (0)

    TDM_ISSUE(0, 0);
#endif

    const int NIT = KCHUNK / TROWS;             // 32
    for (int it = 0; it < NIT; ++it) {
        const int buf = it & 1;
#if HAVE_TDM
        if (it + 1 < NIT) {
            // LDS reads of the buffer we are about to overwrite must be done.
            asm volatile("s_wait_dscnt 0" ::: "memory");
            TDM_ISSUE(it + 1, buf ^ 1);
            __builtin_amdgcn_s_wait_tensorcnt(1);   // current buffer landed
        } else {
            __builtin_amdgcn_s_wait_tensorcnt(0);
        }
        asm volatile("" ::: "memory");
        const int tbase = buf * (TROWS * 16);
#endif
        const int kbase = kc * KCHUNK + it * TROWS;
        for (int r = 0; r < TROWS; r += 4) {
            const int k0   = kbase + r + khalf;
            const int lab0 = labels[k0];
            const int lab1 = labels[k0 + 1];
#if HAVE_TDM
            const int lr = r + khalf;
            const float f0 = smem[tbase + lr * 16 + l15];
            const float f1 = smem[tbase + (lr + 1) * 16 + l15];
#else
            const float f0 = feat[(size_t)k0 * DIMS + ncol];
            const float f1 = feat[((size_t)k0 + 1) * DIMS + ncol];
#endif
            v2f b  = {f0, f1};
            v2f b2 = {f0 * f0, f1 * f1};
#pragma unroll
            for (int t = 0; t < 4; ++t) {
                const int mrow = t * 16 + l15;
                v2f a = {(lab0 == mrow) ? 1.0f : 0.0f,
                         (lab1 == mrow) ? 1.0f : 0.0f};
                accS[t] = __builtin_amdgcn_wmma_f32_16x16x4_f32(
                    false, a, false, b, (short)0, accS[t], false, false);
                accQ[t] = __builtin_amdgcn_wmma_f32_16x16x4_f32(
                    false, a, false, b2, (short)0, accQ[t], false, false);
            }
        }
    }

    // C/D layout: VGPR r, lanes 0-15 -> M=r, N=lane; lanes 16-31 -> M=r+8
    const int mhalf = (lane < 16) ? 0 : 8;
#pragma unroll
    for (int t = 0; t < 4; ++t) {
#pragma unroll
        for (int r = 0; r < 8; ++r) {
            const int m = t * 16 + mhalf + r;
            atomicAdd(&sums [m * DIMS + ncol], accS[t][r]);
            atomicAdd(&sumsq[m * DIMS + ncol], accQ[t][r]);
        }
    }
}

// ---------------- EMA / finalize ----------------
__global__ void fds_finalize(const float* __restrict__ ws,
                             const float* __restrict__ run_mean,
                             const float* __restrict__ run_var,
                             const float* __restrict__ num_tracked,
                             float* __restrict__ out) {
    int i = blockIdx.x * blockDim.x + threadIdx.x;
    if (i >= NBINS * DIMS) return;
    const int b = i / DIMS;
    const int d = i % DIMS;
    const float* cnt = ws;
    const float* sums = ws + MPAD;
    const float* sumsq = ws + MPAD + MPAD * DIMS;
    const float c = cnt[b];
    const float safe_n = fmaxf(c, 1.0f);
    const float mean = sums[b * DIMS + d] / safe_n;
    const float var = (sumsq[b * DIMS + d] - safe_n * mean * mean) /
                      fmaxf(c - 1.0f, 1.0f);
    const bool present = (c > 0.0f);
    const float rm = run_mean[i];
    const float rv = run_var[i];
    out[OUT_NM + i] = present ? (0.1f * mean + 0.9f * rm) : rm;
    out[OUT_NV + i] = present ? (0.1f * var + 0.9f * rv) : rv;
    if (i < NBINS) out[OUT_NN + i] = num_tracked[i] + cnt[i];
}

// ---------------- 5-tap reflect-pad smoothing over bins ----------------
__global__ void fds_smooth(const float* __restrict__ kw, float* __restrict__ out) {
    int i = blockIdx.x * blockDim.x + threadIdx.x;
    if (i >= NBINS * DIMS) return;
    const int b = i / DIMS;
    const int d = i % DIMS;
    float sm = 0.0f, sv = 0.0f;
#pragma unroll
    for (int k = 0; k < 5; ++k) {
        int j = b + k - 2;
        j = (j < 0) ? -j : j;
        j = (j >= NBINS) ? (2 * NBINS - 2 - j) : j;
        const float w = kw[k];
        sm += w * out[OUT_NM + j * DIMS + d];
        sv += w * out[OUT_NV + j * DIMS + d];
    }
    out[OUT_SM + i] = sm;
    out[OUT_SV + i] = sv;
}

extern "C" void kernel_launch(void* const* d_in, const int* in_sizes, int n_in,
                              void* d_out, int out_size, void* d_ws, size_t ws_size,
                              hipStream_t stream) {
    const float* features    = (const float*)d_in[0];
    const int*   labels      = (const int*)d_in[1];
    const float* run_mean    = (const float*)d_in[2];
    const float* run_var     = (const float*)d_in[3];
    const float* num_tracked = (const float*)d_in[4];
    const float* kw          = (const float*)d_in[5];
    float* ws  = (float*)d_ws;
    float* out = (float*)d_out;

    // ws layout: [0,64) counts | [64, 64+64*512) sums | [.., +64*512) sumsq
    const int wsn = MPAD + 2 * MPAD * DIMS;
    fds_zero<<<(wsn + 255) / 256, 256, 0, stream>>>(ws, wsn);
    fds_counts<<<256, 256, 0, stream>>>(labels, ws);

    dim3 grid(DIMS / 16, NTOT / KCHUNK);   // (32 N-tiles, 64 K-chunks)
    fds_wmma_reduce<<<grid, 32, 0, stream>>>(features, labels,
                                             ws + MPAD, ws + MPAD + MPAD * DIMS);

    fds_finalize<<<(NBINS * DIMS + 255) / 256, 256, 0, stream>>>(
        ws, run_mean, run_var, num_tracked, out);
    fds_smooth<<<(NBINS * DIMS + 255) / 256, 256, 0, stream>>>(kw, out);
}